// ForwardDecoder_16956530885118
// MI455X (gfx1250) — compile-verified
//
#include <hip/hip_runtime.h>
#include <stdint.h>

// Pair-HMM forward DP (smoothed max / logsumexp semiring), anti-diagonal
// wavefront, single persistent workgroup on one WGP.
//
// MI455X reasoning: 3071 serial wavefront steps -> latency bound.
//  - hardware s_barrier between steps (one WGP, 32 wave32 waves)
//  - V diagonals in LDS (3 rotating buffers)
//  - theta staged per-diagonal by the Tensor Data Mover (strided 2D gather,
//    36B rows x up-to-1536 rows), double buffered, s_wait_tensorcnt fenced.

#define NN 1536
#define MM 1536
#define NEG_INF (-1.0e8f)
#define NTHREADS 1024

typedef __attribute__((ext_vector_type(4))) unsigned int v4u;
typedef __attribute__((ext_vector_type(8))) int v8i;
typedef __attribute__((ext_vector_type(4))) int v4i;

__device__ __forceinline__ float lse3(float a, float b, float c) {
    float m = fmaxf(a, fmaxf(b, c));
    return m + __logf(__expf(a - m) + __expf(b - m) + __expf(c - m));
}

// Issue one TDM 2D gather: len rows of 9 f32, row stride (MM-1)*9 elements,
// packed contiguously into LDS at lds_byte_off.
__device__ __forceinline__ void tdm_load_diag(const float* theta, int d,
                                              unsigned lds_byte_off) {
    const int ilo = (d - MM > 1) ? (d - MM) : 1;
    const int ihi = (d - 1 < NN) ? (d - 1) : NN;
    const unsigned len = (unsigned)(ihi - ilo + 1);
    const unsigned long long ga =
        (unsigned long long)(uintptr_t)(theta +
            ((unsigned long long)(ilo - 1) * MM + (unsigned)(d - ilo - 1)) * 9ull);

    // D# group 0 (128b): count=1 | lds_addr | global_addr[56:0] | type=2
    v4u g0;
    g0.x = 1u;                                   // count=1, user desc
    g0.y = lds_byte_off;                         // LDS byte address
    g0.z = (unsigned)ga;                         // global addr [31:0]
    g0.w = (unsigned)((ga >> 32) & 0x1FFFFFFull) // global addr [56:32]
           | (2u << 30);                         // type = 2 ("image")
    // D# group 1 (256b)
    v8i g1;
    g1.s0 = (int)(2u << 16);          // wg_mask=0, data_size=2 (4 bytes)
    g1.s1 = (int)(9u << 16);          // atomic_barrier_addr=0, tensor_dim0.lo16=9
    g1.s2 = (int)(len << 16);         // tensor_dim0.hi16=0, tensor_dim1.lo16=len
    g1.s3 = (int)(9u << 16);          // tensor_dim1.hi16=0, tile_dim0=9
    g1.s4 = (int)len;                 // tile_dim1=len, tile_dim2=0
    g1.s5 = (int)((MM - 1) * 9);      // tensor_dim0_stride[31:0] = 13815
    g1.s6 = 0;                        // stride hi bits = 0
    g1.s7 = 0;
    v4i z4 = {0, 0, 0, 0};            // groups 2/3 unused (2D tensor)
    v8i z8 = {0, 0, 0, 0, 0, 0, 0, 0};
    __builtin_amdgcn_tensor_load_to_lds(g0, g1, z4, z4, z8, 0);
}

__global__ __launch_bounds__(NTHREADS) void hmm_forward_wavefront(
        const float* __restrict__ theta, float* __restrict__ out) {
    // theta staging: double buffer, 1536 cells * 9 f32 each = 55296 B per buf
    __shared__ float thbuf[2][1536 * 9];
    // V diagonals: 3 rotating buffers of (NN+1) slots * 3 states
    __shared__ float Vb[3][(NN + 1) * 3];

    const int tid = threadIdx.x;

    // init: diag d=0 (V[0,0,:]=0, rest -inf) and diag d=1 (all -inf)
    for (int t = tid; t < (NN + 1) * 3; t += NTHREADS) {
        Vb[0][t] = NEG_INF;
        Vb[1][t] = NEG_INF;
    }
    if (tid == 0) { Vb[0][0] = 0.0f; Vb[0][1] = 0.0f; Vb[0][2] = 0.0f; }
    __syncthreads();

    // prefetch theta for the first computed diagonal (d=2) into buf 0
    if (tid == 0) {
        tdm_load_diag(theta, 2, (unsigned)(uintptr_t)&thbuf[0][0]);
    }

    int ra = 0, rb = 1, rc = 2;   // ra = diag d-2, rb = diag d-1, rc = diag d
    for (int d = 2; d <= NN + MM; ++d) {
        // fence this step's TDM load, then make it visible workgroup-wide
        if (tid == 0) __builtin_amdgcn_s_wait_tensorcnt(0);
        __syncthreads();
        // prefetch next diagonal into the other buffer (overlaps compute)
        if (tid == 0 && d < NN + MM) {
            tdm_load_diag(theta, d + 1,
                          (unsigned)(uintptr_t)&thbuf[(d + 1) & 1][0]);
        }

        const int ilo = (d - MM > 1) ? (d - MM) : 1;
        const int ihi = (d - 1 < NN) ? (d - 1) : NN;
        const int len = ihi - ilo + 1;
        const int par = d & 1;

        for (int idx = tid; idx < len; idx += NTHREADS) {
            const int i = ilo + idx;           // row; col j = d - i
            const int tb = idx * 9;
            // state 0 (match): diag predecessor V[i-1,j-1] on diag d-2
            float v0 = lse3(Vb[ra][(i - 1) * 3 + 0] + thbuf[par][tb + 0],
                            Vb[ra][(i - 1) * 3 + 1] + thbuf[par][tb + 1],
                            Vb[ra][(i - 1) * 3 + 2] + thbuf[par][tb + 2]);
            // state 1 (X): up predecessor V[i-1,j] on diag d-1
            float v1 = lse3(Vb[rb][(i - 1) * 3 + 0] + thbuf[par][tb + 3],
                            Vb[rb][(i - 1) * 3 + 1] + thbuf[par][tb + 4],
                            Vb[rb][(i - 1) * 3 + 2] + thbuf[par][tb + 5]);
            // state 2 (Y): left predecessor V[i,j-1] on diag d-1
            float v2 = lse3(Vb[rb][i * 3 + 0] + thbuf[par][tb + 6],
                            Vb[rb][i * 3 + 1] + thbuf[par][tb + 7],
                            Vb[rb][i * 3 + 2] + thbuf[par][tb + 8]);
            Vb[rc][i * 3 + 0] = v0;
            Vb[rc][i * 3 + 1] = v1;
            Vb[rc][i * 3 + 2] = v2;
        }
        // border slots of this diagonal: V[0, d] and V[d, 0] are -inf
        if (tid == 0) {
            Vb[rc][0] = NEG_INF; Vb[rc][1] = NEG_INF; Vb[rc][2] = NEG_INF;
            if (d <= NN) {
                Vb[rc][d * 3 + 0] = NEG_INF;
                Vb[rc][d * 3 + 1] = NEG_INF;
                Vb[rc][d * 3 + 2] = NEG_INF;
            }
        }
        __syncthreads();           // publish diag d before rotating
        int t = ra; ra = rb; rb = rc; rc = t;
    }

    // terminal smoothed max over V[N, M, :]; last diag (d=N+M) is in rb
    if (tid == 0) {
        out[0] = lse3(Vb[rb][NN * 3 + 0], Vb[rb][NN * 3 + 1],
                      Vb[rb][NN * 3 + 2]);
    }
}

extern "C" void kernel_launch(void* const* d_in, const int* in_sizes, int n_in,
                              void* d_out, int out_size, void* d_ws,
                              size_t ws_size, hipStream_t stream) {
    (void)in_sizes; (void)n_in; (void)out_size; (void)d_ws; (void)ws_size;
    const float* theta = (const float*)d_in[0];
    float* out = (float*)d_out;
    hipLaunchKernelGGL(hmm_forward_wavefront, dim3(1), dim3(NTHREADS), 0,
                       stream, theta, out);
}